// Network_81947976008063
// MI455X (gfx1250) — compile-verified
//
#include <hip/hip_runtime.h>

// Problem constants (from reference)
#define Bn   2
#define Fn   4
#define Tn   1024
#define Cn   1536
#define NXn  1536
#define NW0  476
#define NW1  476
#define NW2  292
#define NW_PER_I (NW0 + NW1 + NW2)   // 1244
#define E_TOT    (2 * NW_PER_I)      // 2488
#define NT   4                       // t-values per block in main kernel

typedef __attribute__((ext_vector_type(2))) float v2f;
typedef __attribute__((ext_vector_type(8))) float v8f;

// ---------------------------------------------------------------------------
// Prep: build deterministic CSR of (src_channel, mask_weight) per dst channel,
// plus cy[c] (dst multiplicity) and row offsets. Single block, all in LDS.
// ---------------------------------------------------------------------------
__global__ __launch_bounds__(512)
void prep_kernel(const int* __restrict__ wc00, const int* __restrict__ wc01,
                 const int* __restrict__ wc02, const int* __restrict__ wc10,
                 const int* __restrict__ wc11, const int* __restrict__ wc12,
                 const int* __restrict__ xm0,  const int* __restrict__ xm1,
                 int* __restrict__ g_row_off, int* __restrict__ g_cy,
                 int* __restrict__ g_csr_src, float* __restrict__ g_csr_w)
{
    __shared__ int           s_inv[E_TOT];
    __shared__ unsigned char s_flag[E_TOT];
    __shared__ short         s_dst[E_TOT];
    __shared__ short         s_src[E_TOT];
    __shared__ unsigned char s_w[E_TOT];
    __shared__ int           s_cnt[Cn];
    __shared__ int           s_roff[Cn + 1];
    __shared__ int           s_part[512];

    const int tid = threadIdx.x;
    const int nt  = 512;
    const int* wcs[6] = {wc00, wc01, wc02, wc10, wc11, wc12};
    const int joff[3] = {0, NW0, NW0 + NW1};
    const int nwj[3]  = {NW0, NW1, NW2};

    // Phase A: zero
    for (int i = tid; i < E_TOT; i += nt) s_flag[i] = 0;
    for (int i = tid; i < Cn; i += nt)    s_cnt[i]  = 0;
    __syncthreads();

    // Phase B: inverse permutation of ident, and mask flags from xm columns
    for (int i = 0; i < 2; ++i) {
        for (int j = 0; j < 3; ++j) {
            const int* wc = wcs[i * 3 + j];
            const int  o  = i * NW_PER_I + joff[j];
            for (int k = tid; k < nwj[j]; k += nt)
                s_inv[o + wc[2 * k]] = k;        // ident is a permutation
        }
    }
    for (int i = 0; i < 2; ++i) {
        const int* xm = i ? xm1 : xm0;
        for (int m = tid; m < NXn; m += nt) {
            for (int j = 0; j < 3; ++j)
                s_flag[i * NW_PER_I + joff[j] + xm[3 * m + j]] = 1; // idempotent
        }
    }
    __syncthreads();

    // Phase C: per entry k: dst = chan[k], src = chan[inv_ident[k]], w = mask
    for (int e = tid; e < E_TOT; e += nt) {
        const int  i  = e / NW_PER_I;
        const int  r  = e - i * NW_PER_I;
        const int  j  = (r < NW0) ? 0 : (r < NW0 + NW1 ? 1 : 2);
        const int  k  = r - joff[j];
        const int* wc = wcs[i * 3 + j];
        const int  o  = i * NW_PER_I + joff[j];
        const int  dst = wc[2 * k + 1];
        const int  kin = s_inv[o + k];
        s_dst[e] = (short)dst;
        s_src[e] = (short)wc[2 * kin + 1];
        s_w[e]   = s_flag[o + k];
        atomicAdd(&s_cnt[dst], 1);              // integer LDS atomic: deterministic
    }
    __syncthreads();

    // Phase D: exclusive scan over s_cnt -> s_roff  (512 threads x 3 elements)
    const int a0 = s_cnt[3 * tid], a1 = s_cnt[3 * tid + 1], a2 = s_cnt[3 * tid + 2];
    s_part[tid] = a0 + a1 + a2;
    __syncthreads();
    for (int off = 1; off < 512; off <<= 1) {   // inclusive Hillis-Steele
        const int v = s_part[tid];
        const int u = (tid >= off) ? s_part[tid - off] : 0;
        __syncthreads();
        s_part[tid] = v + u;
        __syncthreads();
    }
    const int base = (tid > 0) ? s_part[tid - 1] : 0;
    s_roff[3 * tid]     = base;
    s_roff[3 * tid + 1] = base + a0;
    s_roff[3 * tid + 2] = base + a0 + a1;
    if (tid == 511) s_roff[Cn] = s_part[511];
    __syncthreads();

    // Phase E: deterministic rank (entry order) -> CSR position; write to global
    for (int e = tid; e < E_TOT; e += nt) {
        const short d = s_dst[e];
        int rank = 0;
        for (int e2 = 0; e2 < e; ++e2) rank += (s_dst[e2] == d) ? 1 : 0;
        const int p = s_roff[d] + rank;
        g_csr_src[p] = (int)s_src[e];
        g_csr_w[p]   = s_w[e] ? 1.0f : 0.0f;
    }
    for (int c = tid; c <= Cn; c += nt) g_row_off[c] = s_roff[c];
    for (int c = tid; c < Cn; c += nt)  g_cy[c]      = s_cnt[c];
}

// ---------------------------------------------------------------------------
// Main fused pass. Block = (b, 4 consecutive t). Phase 1: xw = sum_f w[f]*x via
// V_WMMA_F32_16X16X4_F32 (A = 16 channels x 4 f, B = w_seg rows) into LDS.
// Phase 2: CSR gather per channel, scale by 1/max(cy,1), sigmoid, float4 store.
// ---------------------------------------------------------------------------
__global__ __launch_bounds__(256)
void mix_kernel(const float* __restrict__ x, const float* __restrict__ wseg,
                const float* __restrict__ bseg,
                const int* __restrict__ row_off, const int* __restrict__ cy,
                const int* __restrict__ csr_src, const float* __restrict__ csr_w,
                float* __restrict__ out)
{
    __shared__ float s_xw[NT][Cn];

    const int tid = threadIdx.x;
    const int b   = blockIdx.x >> 8;            // Tn/NT = 256 tiles per batch
    const int t0  = (blockIdx.x & 255) * NT;

    const float w0 = wseg[0], w1 = wseg[1], w2 = wseg[2], w3 = wseg[3];
    const float bv = bseg[0];

    const int lane = tid & 31;
    const int wv   = tid >> 5;                  // 8 waves of 32
    const int m    = lane & 15;                 // M within tile
    const int hi   = lane >> 4;                 // K-half select

    // B matrix (4x16): row K broadcast of w_seg[K]; lanes 0-15 rows K=0,2 / 16-31 rows K=1,3
    v2f bmat;
    bmat.x = hi ? w1 : w0;
    bmat.y = hi ? w3 : w2;

    const int TILES = NT * (Cn / 16);           // 384 tiles of 16 channels
    for (int tile = wv; tile < TILES; tile += 8) {
        const int tl = tile / (Cn / 16);
        const int c0 = (tile - tl * (Cn / 16)) * 16;
        const int t  = t0 + tl;

        // A matrix (16x4): lane m holds x[b, 2*hi + {0,1}, t, c0+m]
        const size_t base = (((size_t)b * Fn + 2 * hi) * Tn + t) * (size_t)Cn + c0 + m;
        v2f a;
        a.x = x[base];
        a.y = x[base + (size_t)Tn * Cn];

        // prefetch next tile's A data (-> global_prefetch_b8)
        const int ntile = tile + 8;
        if (ntile < TILES) {
            const int ntl = ntile / (Cn / 16);
            const int nc0 = (ntile - ntl * (Cn / 16)) * 16;
            __builtin_prefetch(
                &x[(((size_t)b * Fn + 2 * hi) * Tn + (t0 + ntl)) * (size_t)Cn + nc0 + m], 0, 1);
        }

        v8f c8 = {};
        // D[m,n] = sum_f x[f, c0+m] * w_seg[f]  (replicated across N)
        v8f d = __builtin_amdgcn_wmma_f32_16x16x4_f32(
            /*neg_a=*/false, a, /*neg_b=*/false, bmat,
            /*c_mod=*/(short)0, c8, /*reuse_a=*/false, /*reuse_b=*/false);

        if (m == 0) {                            // lanes 0 and 16 hold M=0..7 / M=8..15
            float* p = &s_xw[tl][c0 + 8 * hi];
            *reinterpret_cast<float4*>(p)     = make_float4(d[0], d[1], d[2], d[3]);
            *reinterpret_cast<float4*>(p + 4) = make_float4(d[4], d[5], d[6], d[7]);
        }
    }
    __syncthreads();

    // Phase 2: per output channel, gather CSR row for all 4 t's, sigmoid, store float4
    for (int c = tid; c < Cn; c += 256) {
        const int s = row_off[c], e = row_off[c + 1];
        float acc0 = 0.f, acc1 = 0.f, acc2 = 0.f, acc3 = 0.f;
        for (int p = s; p < e; ++p) {
            const int   sc = csr_src[p];
            const float w  = csr_w[p];
            acc0 += s_xw[0][sc] * w;
            acc1 += s_xw[1][sc] * w;
            acc2 += s_xw[2][sc] * w;
            acc3 += s_xw[3][sc] * w;
        }
        const float inv = 1.0f / fmaxf((float)cy[c], 1.0f);
        float4 o;
        o.x = 1.0f / (1.0f + __expf(-(bv + acc0 * inv)));
        o.y = 1.0f / (1.0f + __expf(-(bv + acc1 * inv)));
        o.z = 1.0f / (1.0f + __expf(-(bv + acc2 * inv)));
        o.w = 1.0f / (1.0f + __expf(-(bv + acc3 * inv)));
        *reinterpret_cast<float4*>(out + ((size_t)b * Cn + c) * Tn + t0) = o;
    }
}

// ---------------------------------------------------------------------------
extern "C" void kernel_launch(void* const* d_in, const int* in_sizes, int n_in,
                              void* d_out, int out_size, void* d_ws, size_t ws_size,
                              hipStream_t stream)
{
    const float* x    = (const float*)d_in[0];
    const int*   wc00 = (const int*)d_in[1];
    const int*   wc01 = (const int*)d_in[2];
    const int*   wc02 = (const int*)d_in[3];
    const int*   wc10 = (const int*)d_in[4];
    const int*   wc11 = (const int*)d_in[5];
    const int*   wc12 = (const int*)d_in[6];
    const int*   xm0  = (const int*)d_in[7];
    const int*   xm1  = (const int*)d_in[8];
    const float* wseg = (const float*)d_in[9];
    const float* bseg = (const float*)d_in[10];

    // Workspace layout (~32 KB)
    int*   row_off = (int*)d_ws;               // Cn+1
    int*   cy      = row_off + (Cn + 1);       // Cn
    int*   csr_src = cy + Cn;                  // E_TOT
    float* csr_w   = (float*)(csr_src + E_TOT);// E_TOT

    prep_kernel<<<1, 512, 0, stream>>>(wc00, wc01, wc02, wc10, wc11, wc12,
                                       xm0, xm1, row_off, cy, csr_src, csr_w);

    mix_kernel<<<Bn * (Tn / NT), 256, 0, stream>>>(
        x, wseg, bseg, row_off, cy, csr_src, csr_w, (float*)d_out);
}